// TreeModel_11673721110706
// MI455X (gfx1250) — compile-verified
//
#include <hip/hip_runtime.h>

typedef __attribute__((ext_vector_type(2))) float v2f;
typedef __attribute__((ext_vector_type(8))) float v8f;

#define N_LEAVES 16384
#define DIM      512
#define N_NODES  (2 * N_LEAVES - 1)   // 32767
#define N_TILES  (N_LEAVES / 16)      // 1024 tiles of 16 leaves
#define SQ_WAVES 2048                 // waves in sumsq_kernel (256 blocks x 8)

// ---------------------------------------------------------------------------
// Kernel 1: per-wave partials of
//   delta_total = sq[0] + sum_{n>=1} sq[n] / max(h[n]-h[parent], 1e-7)
// One wave per node row (grid-stride, fixed order => deterministic);
// 16 floats/lane via float4 loads; wave32 shfl_xor reduction;
// one partial per wave written to workspace.
// ---------------------------------------------------------------------------
__global__ __launch_bounds__(256) void sumsq_kernel(const float* __restrict__ deltas,
                                                    const float* __restrict__ heights,
                                                    float* __restrict__ partials) {
    const int lane   = threadIdx.x & 31;
    const int wave   = blockIdx.x * (blockDim.x >> 5) + (threadIdx.x >> 5);
    const int nWaves = gridDim.x * (blockDim.x >> 5);

    float acc = 0.0f;
    for (int n = wave; n < N_NODES; n += nWaves) {
        const float4* row = (const float4*)(deltas + (size_t)n * DIM);
        float s = 0.0f;
#pragma unroll
        for (int it = 0; it < 4; ++it) {
            float4 v = row[it * 32 + lane];
            s += v.x * v.x + v.y * v.y + v.z * v.z + v.w * v.w;
        }
#pragma unroll
        for (int off = 16; off; off >>= 1) s += __shfl_xor(s, off, 32);
        if (lane == 0) {
            if (n == 0) {
                acc += s;
            } else {
                float b = heights[n] - heights[(n - 1) >> 1];
                b = fmaxf(b, 1e-7f);
                acc += s / b;
            }
        }
    }
    if (lane == 0) partials[wave] = acc;
}

// ---------------------------------------------------------------------------
// Kernel 1b: deterministic final reduction of SQ_WAVES partials -> out[16384].
// Single block, fixed summation order.
// ---------------------------------------------------------------------------
__global__ __launch_bounds__(256) void reduce_total_kernel(const float* __restrict__ partials,
                                                           float* __restrict__ out) {
    __shared__ float red[256];
    const int tid = threadIdx.x;
    float s = 0.0f;
#pragma unroll
    for (int i = 0; i < SQ_WAVES / 256; ++i) s += partials[tid + i * 256];
    red[tid] = s;
    __syncthreads();
#pragma unroll
    for (int off = 128; off; off >>= 1) {
        if (tid < off) red[tid] += red[tid + off];
        __syncthreads();
    }
    if (tid == 0) out[N_LEAVES] = red[0];
}

// ---------------------------------------------------------------------------
// Kernel 2: W10[t] = sum of deltas along root..level-10 node (1023+t), t<1024.
// One block per level-10 node; 256 threads x 2 columns. Upper-level rows are
// tiny (2 MB total) and L2-resident across blocks.
// ---------------------------------------------------------------------------
__global__ __launch_bounds__(256) void w10_kernel(const float* __restrict__ deltas,
                                                  float* __restrict__ w10) {
    const int t = blockIdx.x;           // 0..1023
    const int c = threadIdx.x * 2;      // column pair
    int v = 1023 + t;                   // level-10 node
    float2 acc = make_float2(0.0f, 0.0f);
#pragma unroll
    for (int l = 0; l < 11; ++l) {      // levels 10 down to 0
        const float2 d = *(const float2*)(deltas + (size_t)v * DIM + c);
        acc.x += d.x;
        acc.y += d.y;
        v = (v - 1) >> 1;
    }
    *(float2*)(w10 + (size_t)t * DIM + c) = acc;
}

// ---------------------------------------------------------------------------
// Kernel 3: per-tile dense GEMM C(16x32) = X_tile(16x512) * D_tile^T via
// V_WMMA_F32_16X16X4_F32 chains (two 16x16 accumulators). K is split across
// 4 waves per tile (128 each) for latency hiding; partial C tiles are summed
// during the per-leaf diagonal extraction:
//   result[m] = C[m, m>>3] + C[m, 2+(m>>2)] + C[m, 6+(m>>1)] + C[m, 14+m] + C[m, 30]
// Columns: 0-1 = level-11 nodes, 2-5 = level-12, 6-13 = level-13,
//          14-29 = level-14 (leaf deltas), 30 = W10 prefix, 31 = pad (unused).
//
// Operand staging per ISA layouts (wave32):
//   A 16x4 f32 : lane L holds A[L&15, kb+2*(L>>4) .. +1]  -> one b64 load
//   B 4x16 f32 : lane L holds B[kb+2*(L>>4).., N=L&15]   -> one b64 load of
//                D-row (L&15) at the same column offset (B = D^T)
// ---------------------------------------------------------------------------
__global__ __launch_bounds__(256) void dot_wmma_kernel(const float* __restrict__ x,
                                                       const float* __restrict__ deltas,
                                                       const float* __restrict__ w10,
                                                       float* __restrict__ out) {
    __shared__ float lds[8 * 512];                 // 8 waves x (16x32) partial C
    const int lane      = threadIdx.x & 31;
    const int waveInBlk = threadIdx.x >> 5;        // 0..7
    const int tileInBlk = waveInBlk >> 2;          // 0..1
    const int kslice    = waveInBlk & 3;           // 0..3  (K range of 128)
    const int t         = blockIdx.x * 2 + tileInBlk;   // tile id, 0..1023
    const int n         = lane & 15;               // row (A) / column (B)
    const int hi        = lane >> 4;               // K sub-offset select
    const int k0        = kslice * (DIM / 4);      // this wave's K base
    const int k1        = k0 + (DIM / 4);

    // A operand pointer: x row for leaf (16*t + n), K offset 2*hi.
    const float* aPtr = x + (size_t)(16 * t + n) * DIM + 2 * hi;

    // B0 (columns 0..15): ancestor delta rows.
    int node0;
    if (n < 2)       node0 = 2047  + 2  * t + n;          // level 11
    else if (n < 6)  node0 = 4095  + 4  * t + (n - 2);    // level 12
    else if (n < 14) node0 = 8191  + 8  * t + (n - 6);    // level 13
    else             node0 = 16383 + 16 * t + (n - 14);   // level 14 (leaves 0,1)
    const float* b0Ptr = deltas + (size_t)node0 * DIM + 2 * hi;

    // B1 (columns 16..31): leaves 2..15, then W10 prefix, col 31 = dummy.
    const int col1 = 16 + n;
    const float* b1Ptr;
    if (col1 < 30) b1Ptr = deltas + (size_t)(16383 + 16 * t + (col1 - 14)) * DIM + 2 * hi;
    else           b1Ptr = w10 + (size_t)t * DIM + 2 * hi;   // col 30 real, col 31 never read

    v8f c0 = {};
    v8f c1 = {};
#pragma unroll 4
    for (int kb = k0; kb < k1; kb += 4) {
        v2f a  = *(const v2f*)(aPtr  + kb);
        v2f b0 = *(const v2f*)(b0Ptr + kb);
        v2f b1 = *(const v2f*)(b1Ptr + kb);
        c0 = __builtin_amdgcn_wmma_f32_16x16x4_f32(false, a, false, b0, (short)0, c0, false, false);
        c1 = __builtin_amdgcn_wmma_f32_16x16x4_f32(false, a, false, b1, (short)0, c1, false, false);
    }

    // Dump partial C to LDS as [m][col]:
    // VGPR v, lane L -> row v+8*(L>>4), col L&15 (+16 for c1).
    float* my = lds + waveInBlk * 512;
#pragma unroll
    for (int v = 0; v < 8; ++v) {
        my[(v + 8 * hi) * 32 + n]      = c0[v];
        my[(v + 8 * hi) * 32 + 16 + n] = c1[v];
    }
    __syncthreads();

    // One wave per tile extracts; sums the 4 K-slice partials per entry.
    if (kslice == 0 && lane < 16) {
        const int m = lane;
        float res = 0.0f;
#pragma unroll
        for (int s = 0; s < 4; ++s) {
            const float* r = lds + (tileInBlk * 4 + s) * 512 + m * 32;
            res += r[m >> 3]            // level-11 ancestor
                 + r[2 + (m >> 2)]      // level-12
                 + r[6 + (m >> 1)]      // level-13
                 + r[14 + m]            // level-14 (own delta)
                 + r[30];               // root..level-10 prefix
        }
        out[16 * t + m] = res;
    }
}

// ---------------------------------------------------------------------------
extern "C" void kernel_launch(void* const* d_in, const int* in_sizes, int n_in,
                              void* d_out, int out_size, void* d_ws, size_t ws_size,
                              hipStream_t stream) {
    (void)in_sizes; (void)n_in; (void)out_size; (void)ws_size;

    const float* x       = (const float*)d_in[0];
    const float* deltas  = (const float*)d_in[1];
    const float* heights = (const float*)d_in[2];
    float*       out     = (float*)d_out;

    // Workspace layout: [0, 2MB) W10 prefix rows; then SQ_WAVES partials.
    float* w10      = (float*)d_ws;                       // 1024*512 floats
    float* partials = (float*)d_ws + (size_t)N_TILES * DIM;

    sumsq_kernel<<<256, 256, 0, stream>>>(deltas, heights, partials);
    reduce_total_kernel<<<1, 256, 0, stream>>>(partials, out);
    w10_kernel<<<N_TILES, 256, 0, stream>>>(deltas, w10);
    dot_wmma_kernel<<<N_TILES / 2, 256, 0, stream>>>(x, deltas, w10, out);
}